// BarrelShifter8_23407571764167
// MI455X (gfx1250) — compile-verified
//
#include <hip/hip_runtime.h>
#include <stdint.h>

// BarrelShifter8: out[row][j] = (j >= k) ? P[row][j-k] : 0,  k = 4*s2 + 2*s1 + s0
// Pure streaming kernel: ~304 MB of traffic -> HBM-bandwidth bound (~13 us @ 23.3 TB/s).
// Uses CDNA5 async global->LDS copies (ASYNCcnt) for the P tile and NT stores for output.

#define THREADS 256

typedef __attribute__((ext_vector_type(4))) float v4f;

__global__ __launch_bounds__(THREADS) void barrel8_kernel(
    const float* __restrict__ P, const float* __restrict__ S,
    float* __restrict__ out, int N)
{
    __shared__ v4f tileP[THREADS * 2];   // 8 KB: 256 rows x 32 bytes

    const int tid = threadIdx.x;
    const long long row = (long long)blockIdx.x * THREADS + tid;
    if (row >= N) return;

    // ---- async global->LDS DMA of this lane's 32-byte P row (CDNA5 path) ----
    // Low 32 bits of a flat LDS pointer are the LDS byte offset (ISA 10.2).
    const uint32_t lds_base = (uint32_t)(uintptr_t)(&tileP[0]);
    const uint32_t lds0 = lds_base + (uint32_t)tid * 32u;
    const uint32_t lds1 = lds0 + 16u;
    const float* gp0 = P + row * 8;
    const float* gp1 = gp0 + 4;
    asm volatile("global_load_async_to_lds_b128 %0, %1, off"
                 :: "v"(lds0), "v"(gp0) : "memory");
    asm volatile("global_load_async_to_lds_b128 %0, %1, off"
                 :: "v"(lds1), "v"(gp1) : "memory");

    // ---- S row (12 bytes, 4-byte aligned only) : three NT dword loads ----
    const float* sp = S + row * 3;
    const float s0 = __builtin_nontemporal_load(sp + 0);
    const float s1 = __builtin_nontemporal_load(sp + 1);
    const float s2 = __builtin_nontemporal_load(sp + 2);

    // Wait for our own async copies; no barrier needed (lane reads only its own bytes,
    // ASYNCcnt is per-wave).
    asm volatile("s_wait_asynccnt 0" ::: "memory");

    const v4f lo = tileP[tid * 2 + 0];
    const v4f hi = tileP[tid * 2 + 1];

    float x[8] = {lo[0], lo[1], lo[2], lo[3], hi[0], hi[1], hi[2], hi[3]};
    float y[8];

    const bool c2 = s2 >= 0.5f;   // stage 1: shift by 4
    const bool c1 = s1 >= 0.5f;   // stage 2: shift by 2
    const bool c0 = s0 >= 0.5f;   // stage 3: shift by 1

#pragma unroll
    for (int j = 0; j < 8; ++j)
        y[j] = c2 ? ((j >= 4) ? x[j - 4] : 0.0f) : x[j];
#pragma unroll
    for (int j = 0; j < 8; ++j)
        x[j] = c1 ? ((j >= 2) ? y[j - 2] : 0.0f) : y[j];
#pragma unroll
    for (int j = 0; j < 8; ++j)
        y[j] = c0 ? ((j >= 1) ? x[j - 1] : 0.0f) : x[j];

    // ---- streamed output: two non-temporal b128 stores ----
    v4f* op = (v4f*)(out + row * 8);
    v4f o0 = {y[0], y[1], y[2], y[3]};
    v4f o1 = {y[4], y[5], y[6], y[7]};
    __builtin_nontemporal_store(o0, op + 0);
    __builtin_nontemporal_store(o1, op + 1);
}

extern "C" void kernel_launch(void* const* d_in, const int* in_sizes, int n_in,
                              void* d_out, int out_size, void* d_ws, size_t ws_size,
                              hipStream_t stream) {
    const float* P = (const float*)d_in[0];   // (N, 8) float32, values in {0,1}
    const float* S = (const float*)d_in[1];   // (N, 3) float32, values in {0,1}
    float* out = (float*)d_out;               // (N, 8) float32

    const int N = in_sizes[0] / 8;
    const int blocks = (N + THREADS - 1) / THREADS;
    barrel8_kernel<<<blocks, THREADS, 0, stream>>>(P, S, out, N);
}